// KGAT_BiInteraction_49727131353584
// MI455X (gfx1250) — compile-verified
//
#include <hip/hip_runtime.h>

#define N_USERS   50000
#define N_ENT     100000
#define NN        150000      // N_NODES
#define DIM       64
#define BATCH     4096
#define NEG_SLOPE 0.01f
#define L2EPS     1e-12f

typedef float v2f __attribute__((ext_vector_type(2)));
typedef float v8f __attribute__((ext_vector_type(8)));

// ---------------------------------------------------------------- zero scratch
__global__ void zero_kernel(float* __restrict__ p, int n) {
    int i = blockIdx.x * blockDim.x + threadIdx.x;
    int stride = gridDim.x * blockDim.x;
    for (; i < n; i += stride) p[i] = 0.0f;
}

// ------------------------------------------------- edge gather + scatter-add
// 16 threads per edge, float4 per thread. unsafeAtomicAdd -> native
// GLOBAL_ATOMIC_ADD_F32; the 38MB accumulator is L2-resident (192MB L2).
__global__ void edge_scatter(const int*   __restrict__ target,
                             const int*   __restrict__ neighbor,
                             const float* __restrict__ values,
                             const float* __restrict__ userE,
                             const float* __restrict__ entE,
                             const float* __restrict__ feat,   // null => layer0 split source
                             float*       __restrict__ hneigh,
                             int nEdges) {
    int gid = blockIdx.x * blockDim.x + threadIdx.x;
    int e = gid >> 4;
    if (e >= nEdges) return;
    int c = (gid & 15) * 4;

    int   t = target[e];
    int   n = neighbor[e];
    float v = values[e];

    const float* row;
    if (feat) row = feat + (size_t)n * DIM;
    else      row = (n < N_USERS) ? userE + (size_t)n * DIM
                                  : entE  + (size_t)(n - N_USERS) * DIM;

    float4 x = *(const float4*)(row + c);
    float* dst = hneigh + (size_t)t * DIM + c;
    unsafeAtomicAdd(dst + 0, x.x * v);
    unsafeAtomicAdd(dst + 1, x.y * v);
    unsafeAtomicAdd(dst + 2, x.z * v);
    unsafeAtomicAdd(dst + 3, x.w * v);
}

// --------------------------------------------------- per-tile epilogue helper
// bias + LeakyReLU + row L2-normalize + store for one 16-row tile set.
template <int NT>
__device__ __forceinline__ void finish_store(v8f (&acc)[NT],
                                             const float* __restrict__ B1,
                                             const float* __restrict__ B2,
                                             int row, int half, int base,
                                             float* __restrict__ out) {
    const int OUT = NT * 16;
#pragma unroll
    for (int t = 0; t < NT; ++t) {
        float bias = B1[t * 16 + row] + B2[t * 16 + row];
#pragma unroll
        for (int r = 0; r < 8; ++r) {
            float x = acc[t][r] + bias;
            acc[t][r] = (x >= 0.0f) ? x : NEG_SLOPE * x;
        }
    }
#pragma unroll
    for (int r = 0; r < 8; ++r) {
        float s = 0.0f;
#pragma unroll
        for (int t = 0; t < NT; ++t) { float x = acc[t][r]; s += x * x; }
        s += __shfl_xor(s, 1, 32);
        s += __shfl_xor(s, 2, 32);
        s += __shfl_xor(s, 4, 32);
        s += __shfl_xor(s, 8, 32);
        float inv = 1.0f / fmaxf(sqrtf(s), L2EPS);
#pragma unroll
        for (int t = 0; t < NT; ++t) acc[t][r] *= inv;
    }
#pragma unroll
    for (int r = 0; r < 8; ++r) {
        int m = base + r + 8 * half;
#pragma unroll
        for (int t = 0; t < NT; ++t)
            out[(size_t)m * OUT + t * 16 + row] = acc[t][r];
    }
}

// ------------------------------------------- dense bi-interaction layer (WMMA)
// W1/W2 staged in LDS once per block (shared by 8 waves); each wave processes
// TWO 16-node tiles -> two independent WMMA accumulator chains per B fragment.
// D = (F+H) @ W1^T + (F*H) @ W2^T via V_WMMA_F32_16X16X4_F32, K=64 step 4.
template <int NT>
__global__ void dense_layer(const float* __restrict__ userE,
                            const float* __restrict__ entE,
                            const float* __restrict__ feat,   // null => layer0 split source
                            const float* __restrict__ hneigh,
                            const float* __restrict__ W1, const float* __restrict__ B1,
                            const float* __restrict__ W2, const float* __restrict__ B2,
                            float* __restrict__ out, int nTiles) {
    constexpr int OUT = NT * 16;
    __shared__ float sW1[OUT * DIM];
    __shared__ float sW2[OUT * DIM];
    for (int i = threadIdx.x; i < OUT * DIM / 4; i += blockDim.x) {
        ((float4*)sW1)[i] = ((const float4*)W1)[i];
        ((float4*)sW2)[i] = ((const float4*)W2)[i];
    }
    __syncthreads();

    int pair  = blockIdx.x * (blockDim.x >> 5) + (threadIdx.x >> 5);
    int tile0 = pair * 2;
    if (tile0 >= nTiles) return;               // wave-uniform: EXEC stays all-ones
    int  tile1 = tile0 + 1;
    bool has1  = tile1 < nTiles;
    int  tile1c = has1 ? tile1 : tile0;        // clamp addresses, discard results

    int lane = threadIdx.x & 31;
    int row  = lane & 15;                      // A: matrix row; B: matrix column
    int half = lane >> 4;
    int koff = half * 2;                       // k-pair owned by this lane half
    int node0 = tile0  * 16 + row;
    int node1 = tile1c * 16 + row;

    auto rowptr = [&](int node) -> const float* {
        return feat ? feat + (size_t)node * DIM
                    : (node < N_USERS ? userE + (size_t)node * DIM
                                      : entE  + (size_t)(node - N_USERS) * DIM);
    };
    const float* f0 = rowptr(node0);
    const float* f1 = rowptr(node1);
    const float* h0 = hneigh + (size_t)node0 * DIM;
    const float* h1 = hneigh + (size_t)node1 * DIM;

    v8f acc0[NT], acc1[NT];
#pragma unroll
    for (int t = 0; t < NT; ++t)
#pragma unroll
        for (int r = 0; r < 8; ++r) { acc0[t][r] = 0.0f; acc1[t][r] = 0.0f; }

    for (int k0 = 0; k0 < DIM; k0 += 4) {
        v2f fa = *(const v2f*)(f0 + k0 + koff);
        v2f ha = *(const v2f*)(h0 + k0 + koff);
        v2f fb = *(const v2f*)(f1 + k0 + koff);
        v2f hb = *(const v2f*)(h1 + k0 + koff);
        v2f a1_0 = fa + ha, a2_0 = fa * ha;    // sum / bi-interaction, tile0
        v2f a1_1 = fb + hb, a2_1 = fb * hb;    // sum / bi-interaction, tile1
#pragma unroll
        for (int t = 0; t < NT; ++t) {
            int bidx = (t * 16 + row) * DIM + koff + k0;
            v2f b1 = *(const v2f*)(sW1 + bidx);
            v2f b2 = *(const v2f*)(sW2 + bidx);
            acc0[t] = __builtin_amdgcn_wmma_f32_16x16x4_f32(
                false, a1_0, false, b1, (short)0, acc0[t], false, false);
            acc1[t] = __builtin_amdgcn_wmma_f32_16x16x4_f32(
                false, a1_1, false, b1, (short)0, acc1[t], false, false);
            acc0[t] = __builtin_amdgcn_wmma_f32_16x16x4_f32(
                false, a2_0, false, b2, (short)0, acc0[t], false, false);
            acc1[t] = __builtin_amdgcn_wmma_f32_16x16x4_f32(
                false, a2_1, false, b2, (short)0, acc1[t], false, false);
        }
    }

    finish_store<NT>(acc0, B1, B2, row, half, tile0 * 16, out);
    if (has1)
        finish_store<NT>(acc1, B1, B2, row, half, tile1 * 16, out);
}

// ------------------------------------------------------------- score kernel
// One wave per (batch, pos/neg) pair; 160-dim dot over the concat embedding.
__global__ void score_kernel(const int* __restrict__ uids,
                             const int* __restrict__ pids,
                             const int* __restrict__ nids,
                             const float* __restrict__ userE,
                             const float* __restrict__ entE,
                             const float* __restrict__ e1,
                             const float* __restrict__ e2,
                             float* __restrict__ out) {
    int wave = blockIdx.x * (blockDim.x >> 5) + (threadIdx.x >> 5);
    if (wave >= BATCH * 2) return;
    int lane  = threadIdx.x & 31;
    int b     = wave >> 1;
    int which = wave & 1;
    int u   = uids[b];
    int it  = (which ? nids : pids)[b];
    int itn = it + N_USERS;

    float s = 0.0f;
    for (int d = lane; d < 160; d += 32) {
        float a, c;
        if (d < 64)       { a = userE[(size_t)u * 64 + d];        c = entE[(size_t)it * 64 + d]; }
        else if (d < 128) { int dd = d - 64;
                            a = e1[(size_t)u * 64 + dd];          c = e1[(size_t)itn * 64 + dd]; }
        else              { int dd = d - 128;
                            a = e2[(size_t)u * 32 + dd];          c = e2[(size_t)itn * 32 + dd]; }
        s += a * c;
    }
    s += __shfl_xor(s, 16, 32);
    s += __shfl_xor(s,  8, 32);
    s += __shfl_xor(s,  4, 32);
    s += __shfl_xor(s,  2, 32);
    s += __shfl_xor(s,  1, 32);
    if (lane == 0) out[which * BATCH + b] = s;
}

// ---------------------------------------------------------------------------
extern "C" void kernel_launch(void* const* d_in, const int* in_sizes, int n_in,
                              void* d_out, int out_size, void* d_ws, size_t ws_size,
                              hipStream_t stream) {
    const int*   target   = (const int*)  d_in[0];
    const int*   neighbor = (const int*)  d_in[1];
    const float* values   = (const float*)d_in[2];
    const int*   user_ids = (const int*)  d_in[3];
    const int*   pos_ids  = (const int*)  d_in[4];
    const int*   neg_ids  = (const int*)  d_in[5];
    const float* userE    = (const float*)d_in[6];
    const float* entE     = (const float*)d_in[7];
    const float* W1_0 = (const float*)d_in[8];
    const float* b1_0 = (const float*)d_in[9];
    const float* W2_0 = (const float*)d_in[10];
    const float* b2_0 = (const float*)d_in[11];
    const float* W1_1 = (const float*)d_in[12];
    const float* b1_1 = (const float*)d_in[13];
    const float* W2_1 = (const float*)d_in[14];
    const float* b2_1 = (const float*)d_in[15];
    const int E = in_sizes[0];

    float* hneigh = (float*)d_ws;                       // NN*64 f32
    float* e1 = hneigh + (size_t)NN * 64;               // NN*64 f32
    float* e2 = e1     + (size_t)NN * 64;               // NN*32 f32

    const int nTiles = NN / 16;                         // 9375
    const int nPairs = (nTiles + 1) / 2;                // 4688
    const int denseBlocks = (nPairs + 7) / 8;           // 8 waves / block
    const int edgeBlocks  = (int)(((long long)E * 16 + 255) / 256);

    // ---- layer 0 (64 -> 64)
    zero_kernel<<<2048, 256, 0, stream>>>(hneigh, NN * 64);
    edge_scatter<<<edgeBlocks, 256, 0, stream>>>(target, neighbor, values,
                                                 userE, entE, nullptr, hneigh, E);
    dense_layer<4><<<denseBlocks, 256, 0, stream>>>(userE, entE, nullptr, hneigh,
                                                    W1_0, b1_0, W2_0, b2_0, e1, nTiles);
    // ---- layer 1 (64 -> 32)
    zero_kernel<<<2048, 256, 0, stream>>>(hneigh, NN * 64);
    edge_scatter<<<edgeBlocks, 256, 0, stream>>>(target, neighbor, values,
                                                 userE, entE, e1, hneigh, E);
    dense_layer<2><<<denseBlocks, 256, 0, stream>>>(userE, entE, e1, hneigh,
                                                    W1_1, b1_1, W2_1, b2_1, e2, nTiles);
    // ---- scores (pos | neg), 8 waves / block
    score_kernel<<<(BATCH * 2) / 8, 256, 0, stream>>>(user_ids, pos_ids, neg_ids,
                                                      userE, entE, e1, e2, (float*)d_out);
}